// BoundaryLoss_57114475102495
// MI455X (gfx1250) — compile-verified
//
#include <hip/hip_runtime.h>
#include <hip/hip_bf16.h>

typedef __attribute__((ext_vector_type(2))) float v2f;
typedef __attribute__((ext_vector_type(8))) float v8f;

#define NSITES (8*128*128)   // 131072
#define DCH    128
#define MMEM   1000
#define CAP_A  100
#define CAP_P  333
#define M_PAD  112           // 7 tiles of 16 (>= 100 anchors)
#define N_PAD  1008          // 63 tiles of 16 (>= 1000 memory rows)

// ---- workspace layout (bytes, 256B-aligned blocks) ----
static constexpr size_t OFF_FLAGS = 0;                                 // 131072 u8
static constexpr size_t OFF_ANC   = 131072;                            // 100 int
static constexpr size_t OFF_POS   = OFF_ANC + 512;                     // 333 int
static constexpr size_t OFF_NEG   = OFF_POS + 1536;                    // 333 int
static constexpr size_t OFF_CNT   = OFF_NEG + 1536;                    // 3 int
static constexpr size_t OFF_AW    = OFF_CNT + 256;                     // 112*128 f32
static constexpr size_t OFF_PW    = OFF_AW + (size_t)M_PAD*DCH*4;      // 1008*128 f32
static constexpr size_t OFF_NW    = OFF_PW + (size_t)N_PAD*DCH*4;      // 1008*128 f32
static constexpr size_t OFF_SP    = OFF_NW + (size_t)N_PAD*DCH*4;      // 112*1008 f32
static constexpr size_t OFF_SN    = OFF_SP + (size_t)M_PAD*N_PAD*4;    // 112*1008 f32
static constexpr size_t OFF_PART  = OFF_SN + (size_t)M_PAD*N_PAD*4;    // 100 f32

// ---- kernel 1: per-site argmax label + category flags ----
__global__ __launch_bounds__(256) void k_classify(const float* __restrict__ preds,
                                                  const int* __restrict__ gts,
                                                  unsigned char* __restrict__ flags) {
  int site = blockIdx.x * 256 + threadIdx.x;
  if (site >= NSITES) return;
  int b = site >> 14, rem = site & 16383, i = rem >> 7, j = rem & 127;
  int y = i * 4, x = j * 4;
  const float* p = preds + (size_t)b * 21u * 262144u + (size_t)y * 512u + (size_t)x;
  float best = p[0]; int lbl = 0;
  #pragma unroll
  for (int c = 1; c < 21; ++c) {
    float v = p[(size_t)c * 262144u];
    if (v > best) { best = v; lbl = c; }   // first-max wins (jnp.argmax semantics)
  }
  int gt = gts[(size_t)b * 262144u + (size_t)y * 512u + (size_t)x];
  bool pm = (lbl != 0);
  bool gm = (gt != 0) && (gt != 255);
  bool ig = (gt == 255);
  unsigned f = 0;
  if (pm && gm)        f |= 1u;   // anchor
  if (gm && !pm)       f |= 2u;   // positive
  if (pm && !gm && !ig) f |= 4u;  // negative
  flags[site] = (unsigned char)f;
}

// ---- kernel 2: single-block scan -> first-k selected site indices + counts ----
__global__ __launch_bounds__(1024) void k_select(const unsigned char* __restrict__ flags,
                                                 int* __restrict__ anc_idx,
                                                 int* __restrict__ pos_idx,
                                                 int* __restrict__ neg_idx,
                                                 int* __restrict__ counts) {
  __shared__ unsigned long long sbuf[1024];
  __shared__ unsigned int base[3];
  const int t = threadIdx.x;
  if (t == 0) { base[0] = 0; base[1] = 0; base[2] = 0; }
  __syncthreads();
  const unsigned long long M21 = 0x1FFFFFull;
  for (int chunk = 0; chunk < NSITES / 1024; ++chunk) {
    int site = chunk * 1024 + t;
    unsigned f = flags[site];
    unsigned long long v = (unsigned long long)(f & 1u)
                         | ((unsigned long long)((f >> 1) & 1u) << 21)
                         | ((unsigned long long)((f >> 2) & 1u) << 42);
    sbuf[t] = v;
    __syncthreads();
    for (int off = 1; off < 1024; off <<= 1) {
      unsigned long long tv = (t >= off) ? sbuf[t - off] : 0ull;
      __syncthreads();
      sbuf[t] += tv;
      __syncthreads();
    }
    unsigned long long excl = sbuf[t] - v;
    unsigned pa = base[0] + (unsigned)(excl & M21);
    unsigned pp = base[1] + (unsigned)((excl >> 21) & M21);
    unsigned pn = base[2] + (unsigned)((excl >> 42) & M21);
    if ((f & 1u) && pa < CAP_A) anc_idx[pa] = site;
    if ((f & 2u) && pp < CAP_P) pos_idx[pp] = site;
    if ((f & 4u) && pn < CAP_P) neg_idx[pn] = site;
    __syncthreads();
    if (t == 1023) {
      unsigned long long tot = sbuf[1023];
      base[0] += (unsigned)(tot & M21);
      base[1] += (unsigned)((tot >> 21) & M21);
      base[2] += (unsigned)((tot >> 42) & M21);
    }
    __syncthreads();
  }
  if (t == 0) {
    counts[0] = (int)(base[0] < CAP_A ? base[0] : CAP_A);
    counts[1] = (int)(base[1] < CAP_P ? base[1] : CAP_P);
    counts[2] = (int)(base[2] < CAP_P ? base[2] : CAP_P);
  }
}

// ---- kernel 3: build normalized padded row matrices A / P / N ----
__global__ __launch_bounds__(128) void k_build(const float* __restrict__ emb,
                                               const float* __restrict__ pos_mem,
                                               const float* __restrict__ neg_mem,
                                               const int* __restrict__ anc_idx,
                                               const int* __restrict__ pos_idx,
                                               const int* __restrict__ neg_idx,
                                               const int* __restrict__ counts,
                                               float* __restrict__ Aw,
                                               float* __restrict__ Pw,
                                               float* __restrict__ Nw) {
  int mat = blockIdx.y;        // 0=A, 1=P, 2=N
  int row = blockIdx.x;
  int t = threadIdx.x;         // channel 0..127
  int nrows = (mat == 0) ? M_PAD : N_PAD;
  if (row >= nrows) return;

  const int*   idx = (mat == 0) ? anc_idx : (mat == 1) ? pos_idx : neg_idx;
  const float* mem = (mat == 1) ? pos_mem : neg_mem;
  float*       out = (mat == 0) ? Aw : (mat == 1) ? Pw : Nw;
  int cnt = counts[mat];
  int capk = (mat == 0) ? CAP_A : CAP_P;

  float x = 0.0f;
  bool from_emb = (row < capk) && (row < cnt);
  if (from_emb) {
    int site = idx[row];
    int b = site >> 14, rem = site & 16383, i = rem >> 7, j = rem & 127;
    x = emb[(((size_t)b * DCH + t) * 128 + i) * 128 + j];
  } else if (mat != 0 && row < MMEM) {
    x = mem[(size_t)row * DCH + t];
  }

  __shared__ float red[128];
  if (from_emb) {  // channel L2-norm, eps 1e-12 (uniform branch per block)
    red[t] = x * x; __syncthreads();
    for (int s = 64; s > 0; s >>= 1) { if (t < s) red[t] += red[t + s]; __syncthreads(); }
    float n1 = sqrtf(red[0]);
    x = x / fmaxf(n1, 1e-12f);
    __syncthreads();
  }
  // unit() with eps 1e-8 (applies to anchors and all memory rows; zeros stay zero)
  red[t] = x * x; __syncthreads();
  for (int s = 64; s > 0; s >>= 1) { if (t < s) red[t] += red[t + s]; __syncthreads(); }
  float n2 = sqrtf(red[0]);
  out[(size_t)row * DCH + t] = x / fmaxf(n2, 1e-8f);
}

// ---- kernel 4: WMMA f32 GEMM  S = A (MxK) * Brows^T (KxN), K=128 ----
// One wave per 16x16 tile; V_WMMA_F32_16X16X4_F32, 32 k-steps.
__global__ __launch_bounds__(32) void k_gemm(const float* __restrict__ Aw,
                                             const float* __restrict__ Pw,
                                             const float* __restrict__ Nw,
                                             float* __restrict__ Spos,
                                             float* __restrict__ Sneg) {
  const float* B = (blockIdx.z == 0) ? Pw : Nw;
  float*       S = (blockIdx.z == 0) ? Spos : Sneg;
  int tn = blockIdx.x, tm = blockIdx.y;
  int lane = threadIdx.x;                 // 0..31
  int lo = lane & 15;
  int hi = lane >> 4;
  // f32 fragment layout: A[m=lane%16][k=2*(lane/16)+v], B[k=2*(lane/16)+v][n=lane%16]
  // With B = rows^T this is the same row-major gather for both operands.
  const float* ap = Aw + ((size_t)(tm * 16 + lo)) * DCH + 2 * hi;
  const float* bp = B  + ((size_t)(tn * 16 + lo)) * DCH + 2 * hi;
  v8f c = {};
  #pragma unroll
  for (int k0 = 0; k0 < DCH; k0 += 4) {
    v2f a = *(const v2f*)(ap + k0);
    v2f b = *(const v2f*)(bp + k0);
    c = __builtin_amdgcn_wmma_f32_16x16x4_f32(false, a, false, b, (short)0, c, false, false);
  }
  int mBase = tm * 16 + hi * 8;
  int n = tn * 16 + lo;
  #pragma unroll
  for (int r = 0; r < 8; ++r)
    S[(size_t)(mBase + r) * N_PAD + n] = c[r];
}

// ---- kernel 5: per-anchor relu triplet reduction ----
__global__ __launch_bounds__(256) void k_loss(const float* __restrict__ Spos,
                                              const float* __restrict__ Sneg,
                                              const int* __restrict__ counts,
                                              float* __restrict__ partials) {
  int i = blockIdx.x;  // 0..99
  __shared__ float ps[MMEM];
  __shared__ float ns[MMEM];
  __shared__ float red[256];
  for (int t = threadIdx.x; t < MMEM; t += 256) {
    ps[t] = Spos[(size_t)i * N_PAD + t];
    ns[t] = Sneg[(size_t)i * N_PAD + t];
  }
  __syncthreads();
  float acc = 0.0f;
  for (int p = threadIdx.x; p < MMEM; p += 256) {
    float xp = ps[p] + 0.2f;
    #pragma unroll 4
    for (int n = 0; n < MMEM; ++n)
      acc += fmaxf(xp - ns[n], 0.0f);
  }
  red[threadIdx.x] = acc; __syncthreads();
  for (int s = 128; s > 0; s >>= 1) {
    if (threadIdx.x < (unsigned)s) red[threadIdx.x] += red[threadIdx.x + s];
    __syncthreads();
  }
  if (threadIdx.x == 0) partials[i] = (i < counts[0]) ? red[0] : 0.0f;
}

// ---- kernel 6: final reduce + scale ----
__global__ __launch_bounds__(128) void k_final(const float* __restrict__ partials,
                                               const int* __restrict__ counts,
                                               float* __restrict__ out) {
  __shared__ float red[128];
  int t = threadIdx.x;
  float a = 0.0f;
  for (int i = t; i < CAP_A; i += 128) a += partials[i];
  red[t] = a; __syncthreads();
  for (int s = 64; s > 0; s >>= 1) { if (t < s) red[t] += red[t + s]; __syncthreads(); }
  if (t == 0) {
    int c = counts[0]; if (c < 1) c = 1;
    out[0] = red[0] / ((float)c * 1.0e6f);
  }
}

extern "C" void kernel_launch(void* const* d_in, const int* in_sizes, int n_in,
                              void* d_out, int out_size, void* d_ws, size_t ws_size,
                              hipStream_t stream) {
  (void)in_sizes; (void)n_in; (void)out_size; (void)ws_size;
  const float* preds    = (const float*)d_in[0];
  const float* emb      = (const float*)d_in[1];
  const int*   gts      = (const int*)  d_in[2];
  const float* pos_mem  = (const float*)d_in[3];
  const float* neg_mem  = (const float*)d_in[4];
  char* ws = (char*)d_ws;

  unsigned char* flags = (unsigned char*)(ws + OFF_FLAGS);
  int*   anc_idx = (int*)(ws + OFF_ANC);
  int*   pos_idx = (int*)(ws + OFF_POS);
  int*   neg_idx = (int*)(ws + OFF_NEG);
  int*   counts  = (int*)(ws + OFF_CNT);
  float* Aw      = (float*)(ws + OFF_AW);
  float* Pw      = (float*)(ws + OFF_PW);
  float* Nw      = (float*)(ws + OFF_NW);
  float* Spos    = (float*)(ws + OFF_SP);
  float* Sneg    = (float*)(ws + OFF_SN);
  float* part    = (float*)(ws + OFF_PART);

  k_classify<<<NSITES / 256, 256, 0, stream>>>(preds, gts, flags);
  k_select<<<1, 1024, 0, stream>>>(flags, anc_idx, pos_idx, neg_idx, counts);
  k_build<<<dim3(N_PAD, 3), 128, 0, stream>>>(emb, pos_mem, neg_mem,
                                              anc_idx, pos_idx, neg_idx, counts,
                                              Aw, Pw, Nw);
  k_gemm<<<dim3(N_PAD / 16, M_PAD / 16, 2), 32, 0, stream>>>(Aw, Pw, Nw, Spos, Sneg);
  k_loss<<<CAP_A, 256, 0, stream>>>(Spos, Sneg, counts, part);
  k_final<<<1, 128, 0, stream>>>(part, counts, (float*)d_out);
}